// EdgeConv_83013127897105
// MI455X (gfx1250) — compile-verified
//
#include <hip/hip_runtime.h>

typedef float v2f __attribute__((ext_vector_type(2)));
typedef float v8f __attribute__((ext_vector_type(8)));

#define N_NODES 50000
#define N_EDGES 800000
#define D 64

#define WAVES_PER_BLOCK 4
#define THREADS (WAVES_PER_BLOCK * 32)

// ---------------- Kernel 1: zero node accumulator (workspace) ----------------
__global__ void zero_nodes_kernel(float* __restrict__ node_x) {
    int i = blockIdx.x * blockDim.x + threadIdx.x;       // float4 granularity
    const int total4 = N_NODES * D / 4;
    if (i < total4) {
        ((float4*)node_x)[i] = make_float4(0.f, 0.f, 0.f, 0.f);
    }
}

// ---------------- Kernel 2: segment_sum scatter (edge_feat -> node_x[dst]) --
__global__ void scatter_edges_kernel(const float* __restrict__ edge_feat,
                                     const int* __restrict__ dst,
                                     float* __restrict__ node_x) {
    long long i = (long long)blockIdx.x * blockDim.x + threadIdx.x; // one per 4 floats
    const long long total = (long long)N_EDGES * (D / 4);
    if (i >= total) return;
    int e = (int)(i >> 4);            // 16 float4 chunks per edge (D=64)
    int c = ((int)i & 15) * 4;
    int row = dst[e];
    float4 v = ((const float4*)edge_feat)[i];
    float* p = node_x + (long long)row * D + c;
    // fp32 hardware atomics -> global_atomic_add_f32 (no-return, STOREcnt)
    unsafeAtomicAdd(p + 0, v.x);
    unsafeAtomicAdd(p + 1, v.y);
    unsafeAtomicAdd(p + 2, v.z);
    unsafeAtomicAdd(p + 3, v.w);
}

// ---------------- Kernel 3: per-edge MLP via V_WMMA_F32_16X16X4_F32 ---------
// One wave handles a tile of 16 edges:  out[16x64] = 0.5*(relu(inp@W1+b1)@W2+b2)
//
// Weights are staged in LDS in "pair-major" layout so a lane's B fragment
// (rows 2p and 2p+1, one column) is a single aligned 8-byte LDS load:
//     wls[p*(2*D) + c*2 + j] = W[2p + j][c]
__global__ __launch_bounds__(THREADS) void edge_mlp_kernel(
    const float* __restrict__ node_x,
    const int* __restrict__ src,
    const int* __restrict__ dst,
    const float* __restrict__ W1,
    const float* __restrict__ b1,
    const float* __restrict__ W2,
    const float* __restrict__ b2,
    float* __restrict__ out)
{
    __shared__ float w1s[D * D];                       // 16 KB, pair-major
    __shared__ float w2s[D * D];                       // 16 KB, pair-major
    __shared__ float hs[WAVES_PER_BLOCK][16 * D];      // 16 KB (per-wave h tile)

    const int tid = threadIdx.x;

    // Stage + transpose weights into pair-major LDS layout.
    // q indexes (row-pair p, 4-col chunk c0): (D/2)*(D/4) = 512 chunks.
    for (int q = tid; q < (D / 2) * (D / 4); q += THREADS) {
        int p  = q >> 4;            // row pair 0..31
        int c0 = (q & 15) * 4;      // column chunk
        float* o1 = w1s + p * (2 * D) + c0 * 2;
        float* o2 = w2s + p * (2 * D) + c0 * 2;
        float4 r0 = *(const float4*)(W1 + (2 * p + 0) * D + c0);
        float4 r1 = *(const float4*)(W1 + (2 * p + 1) * D + c0);
        ((v2f*)o1)[0] = (v2f){r0.x, r1.x};
        ((v2f*)o1)[1] = (v2f){r0.y, r1.y};
        ((v2f*)o1)[2] = (v2f){r0.z, r1.z};
        ((v2f*)o1)[3] = (v2f){r0.w, r1.w};
        float4 s0 = *(const float4*)(W2 + (2 * p + 0) * D + c0);
        float4 s1 = *(const float4*)(W2 + (2 * p + 1) * D + c0);
        ((v2f*)o2)[0] = (v2f){s0.x, s1.x};
        ((v2f*)o2)[1] = (v2f){s0.y, s1.y};
        ((v2f*)o2)[2] = (v2f){s0.z, s1.z};
        ((v2f*)o2)[3] = (v2f){s0.w, s1.w};
    }
    __syncthreads();

    const int wave = tid >> 5;
    const int lane = tid & 31;
    const int half = lane >> 4;   // K-half selector for A/B fragments
    const int col  = lane & 15;   // N (or M for A) within 16

    int tile = blockIdx.x * WAVES_PER_BLOCK + wave;
    const int ntiles = N_EDGES / 16;
    if (tile >= ntiles) tile = ntiles - 1;   // clamp (keeps barrier safe; exact grid anyway)
    const long long ebase = (long long)tile * 16;

    // Per-lane edge row (A-matrix M = lane&15); lanes L and L+16 share a row.
    const int m = col;
    const int e = (int)(ebase + m);
    const int srow = src[e];
    const int drow = dst[e];
    const float* psrc = node_x + (long long)srow * D;
    const float* pdst = node_x + (long long)drow * D;

    // ---- GEMM1 A fragments: inp = node_x[src] + node_x[dst], gathered from L2.
    // f32 16x4 A layout: lane holds A[m][4t + half*2 + j] in vgpr j.
    v2f a[16];
#pragma unroll
    for (int t = 0; t < 16; ++t) {
        int k0 = 4 * t + half * 2;
        v2f s = *(const v2f*)(psrc + k0);
        v2f d = *(const v2f*)(pdst + k0);
        a[t] = s + d;
    }

    // ---- GEMM1: h = inp @ W1   (16 k-steps x 4 n-tiles of 16x16x4 f32 WMMA)
    v8f acc[4] = {};
#pragma unroll
    for (int t = 0; t < 16; ++t) {
        // B fragment pair index: p = 2t + half  -> rows 4t+half*2, 4t+half*2+1
        const float* wrow = w1s + (2 * t + half) * (2 * D);
#pragma unroll
        for (int n = 0; n < 4; ++n) {
            v2f b = *(const v2f*)(wrow + (16 * n + col) * 2);
            acc[n] = __builtin_amdgcn_wmma_f32_16x16x4_f32(
                false, a[t], false, b, (short)0, acc[n], false, false);
        }
    }

    // ---- bias + ReLU, stash h tile in LDS (C layout -> row-major 16x64)
    float* hw = &hs[wave][0];
#pragma unroll
    for (int n = 0; n < 4; ++n) {
        float bias = b1[16 * n + col];
#pragma unroll
        for (int r = 0; r < 8; ++r) {
            float v = acc[n][r] + bias;      // C layout: row = half*8 + r
            v = v > 0.f ? v : 0.f;
            hw[(half * 8 + r) * D + 16 * n + col] = v;
        }
    }
    __syncthreads();

    // ---- GEMM2 A fragments from LDS (C layout -> A layout), aligned b64 loads
#pragma unroll
    for (int t = 0; t < 16; ++t) {
        int k0 = 4 * t + half * 2;
        a[t] = *(const v2f*)(hw + m * D + k0);
    }

    // ---- GEMM2: out = 0.5*(h @ W2 + b2)
    v8f acc2[4] = {};
#pragma unroll
    for (int t = 0; t < 16; ++t) {
        const float* wrow = w2s + (2 * t + half) * (2 * D);
#pragma unroll
        for (int n = 0; n < 4; ++n) {
            v2f b = *(const v2f*)(wrow + (16 * n + col) * 2);
            acc2[n] = __builtin_amdgcn_wmma_f32_16x16x4_f32(
                false, a[t], false, b, (short)0, acc2[n], false, false);
        }
    }

    // ---- epilogue store (coalesced 64B runs per 16-lane group)
#pragma unroll
    for (int n = 0; n < 4; ++n) {
        float bias = b2[16 * n + col];
#pragma unroll
        for (int r = 0; r < 8; ++r) {
            float v = 0.5f * (acc2[n][r] + bias);
            out[(ebase + half * 8 + r) * D + 16 * n + col] = v;
        }
    }
}

// ---------------------------------------------------------------------------
extern "C" void kernel_launch(void* const* d_in, const int* in_sizes, int n_in,
                              void* d_out, int out_size, void* d_ws, size_t ws_size,
                              hipStream_t stream) {
    const float* edge_feat = (const float*)d_in[0];
    const int*   src       = (const int*)d_in[1];
    const int*   dst       = (const int*)d_in[2];
    const float* W1        = (const float*)d_in[3];
    const float* b1        = (const float*)d_in[4];
    const float* W2        = (const float*)d_in[5];
    const float* b2        = (const float*)d_in[6];
    float* out = (float*)d_out;
    float* node_x = (float*)d_ws;              // 50000*64*4 = 12.8 MB scratch

    // 1) zero the segment-sum accumulator (every launch; ws is not re-poisoned)
    {
        int total4 = N_NODES * D / 4;
        int blk = 256;
        int grid = (total4 + blk - 1) / blk;
        zero_nodes_kernel<<<grid, blk, 0, stream>>>(node_x);
    }
    // 2) scatter-add edge features into node_x[dst]
    {
        long long total = (long long)N_EDGES * (D / 4);
        int blk = 256;
        int grid = (int)((total + blk - 1) / blk);
        scatter_edges_kernel<<<grid, blk, 0, stream>>>(edge_feat, dst, node_x);
    }
    // 3) per-edge MLP with fp32 WMMA
    {
        int tiles = N_EDGES / 16;                                   // 50000
        int grid = (tiles + WAVES_PER_BLOCK - 1) / WAVES_PER_BLOCK; // 12500
        edge_mlp_kernel<<<grid, THREADS, 0, stream>>>(node_x, src, dst,
                                                      W1, b1, W2, b2, out);
    }
}